// QNet_64750926955162
// MI455X (gfx1250) — compile-verified
//
#include <hip/hip_runtime.h>
#include <hip/hip_bf16.h>

typedef __attribute__((ext_vector_type(2))) float v2f;
typedef __attribute__((ext_vector_type(8))) float v8f;

namespace {
constexpr int NN     = 45056;                  // nodes
constexpr int EE     = 720896;                 // edges
constexpr int FEA    = 60;
constexpr int HID    = 256;
constexpr int MACH   = 22;
constexpr int BATCH  = NN / MACH;              // 2048
constexpr int CAT    = HID + FEA;              // 316
constexpr int MLP_IN = MACH * CAT;             // 6952
constexpr int CK4    = 32;                     // k4-groups per LDS chunk (128 K, 32 KB)
constexpr float NEG  = 0.01f;
}

__device__ __forceinline__ float lrelu(float v) { return v > 0.0f ? v : v * NEG; }

// ---------------- norm pipeline ----------------
__global__ void k_init_deg(float* deg) {
  int i = blockIdx.x * blockDim.x + threadIdx.x;
  if (i < NN) deg[i] = 1.0f;                    // self-loop weight pre-seeded
}

__global__ void k_deg_accum(const long long* __restrict__ col,
                            const float* __restrict__ ew,
                            float* __restrict__ deg) {
  int e = blockIdx.x * blockDim.x + threadIdx.x;
  if (e < EE) atomicAdd(&deg[(int)col[e]], ew[e]);
}

__global__ void k_dinv(float* deg) {
  int i = blockIdx.x * blockDim.x + threadIdx.x;
  if (i < NN) {
    float d = deg[i];
    deg[i] = d > 0.0f ? rsqrtf(fmaxf(d, 1e-12f)) : 0.0f;
  }
}

__global__ void k_norm(const long long* __restrict__ row,
                       const long long* __restrict__ col,
                       const float* __restrict__ ew,
                       const float* __restrict__ dinv,
                       float* __restrict__ norm) {
  int e = blockIdx.x * blockDim.x + threadIdx.x;
  if (e < EE) norm[e] = dinv[(int)row[e]] * ew[e] * dinv[(int)col[e]];
}

__global__ void k_zero(float4* __restrict__ p, int n4) {
  int i = blockIdx.x * blockDim.x + threadIdx.x;
  if (i < n4) p[i] = make_float4(0.f, 0.f, 0.f, 0.f);
}

// Pack W[K,256] into 64-column panels with groups of 4 k-values interleaved:
//   Wp[ ((p*(K/4) + k4)*64 + n%64)*4 + k%4 ] = W[k*256 + n],  p = n/64
// => a lane's B fragment {W[k..k+1][n]} is one aligned b64 load, and a block's
//    panel stream is contiguous for coalesced float4 -> LDS staging.
__global__ void k_pack_w(const float* __restrict__ W, float* __restrict__ Wp, int K) {
  int idx = blockIdx.x * blockDim.x + threadIdx.x;   // K*256
  if (idx >= K * HID) return;
  int k = idx >> 8;
  int n = idx & 255;
  int p = n >> 6, nl = n & 63;
  int k4 = k >> 2, kk = k & 3;
  Wp[(((size_t)p * (K >> 2) + k4) * 64 + nl) * 4 + kk] = W[idx];
}

// ---------------- WMMA fp32 GEMM:  C[M,256] = act(A[M,K] @ W[K,256] + bias) --
// Block = 8 waves -> 128(M) x 64(N) tile. B panel staged through LDS in 32-KB
// chunks shared by all waves; one wave owns a 16x64 strip with 4 accumulators.
// wave32 fragment layout: A lane holds M=lane%16, K pair {0,1}|{2,3} by lane
// half; B mirrors with N=lane%16; C/D VGPR r = rows r / r+8 across halves.
template <bool HAS_BIAS, bool ACT>
__global__ __launch_bounds__(256) void k_gemm_wmma(
    const float* __restrict__ A, const float* __restrict__ Wp,
    const float* __restrict__ bias, float* __restrict__ C, int K) {
  __shared__ float ldsB[CK4 * 64 * 4];          // 32 KB
  const int tid   = threadIdx.x;
  const int lane  = tid & 31;
  const int wave  = tid >> 5;
  const int mTile = blockIdx.x * 8 + wave;
  const int nBase = blockIdx.y * 64;
  const int half  = lane >> 4;                  // 0 | 1
  const int l16   = lane & 15;
  const int kOff  = half * 2;
  const int K4    = K >> 2;

  const float*  __restrict__ Arow   = A + (size_t)(mTile * 16 + l16) * (size_t)K + kOff;
  const float4* __restrict__ Wpanel = (const float4*)(Wp + (size_t)blockIdx.y * (size_t)K4 * 256);

  v8f acc[4] = {};

  for (int k4b = 0; k4b < K4; k4b += CK4) {
    const int cnt = min(CK4, K4 - k4b);
    const float4* __restrict__ src = Wpanel + (size_t)k4b * 64;
    float4* dst = (float4*)ldsB;
    for (int t = tid; t < cnt * 64; t += 256) dst[t] = src[t];   // coalesced stage
    __syncthreads();
    for (int kk = 0; kk < cnt; ++kk) {
      v2f a = *(const v2f*)(Arow + (size_t)(k4b + kk) * 4);
#pragma unroll
      for (int j = 0; j < 4; ++j) {
        v2f b = *(const v2f*)(&ldsB[(kk * 64 + j * 16 + l16) * 4 + kOff]);
        acc[j] = __builtin_amdgcn_wmma_f32_16x16x4_f32(
            false, a, false, b, (short)0, acc[j], false, false);
      }
    }
    __syncthreads();
  }

#pragma unroll
  for (int j = 0; j < 4; ++j) {
    const int n = nBase + j * 16 + l16;
    const float bv = HAS_BIAS ? bias[n] : 0.0f;
#pragma unroll
    for (int r = 0; r < 8; ++r) {
      const int mRow = mTile * 16 + half * 8 + r;
      float v = acc[j][r] + bv;
      if (ACT) v = lrelu(v);
      C[(size_t)mRow * HID + n] = v;
    }
  }
}

// ---------------- GCN message scatter: out[col] += norm * hw[row] ------------
__global__ __launch_bounds__(HID) void k_scatter(
    const float* __restrict__ hw, const long long* __restrict__ row,
    const long long* __restrict__ col, const float* __restrict__ norm,
    float* __restrict__ out) {
  const int e  = blockIdx.x;                    // uniform -> scalar loads
  const int ch = threadIdx.x;
  const int r  = (int)row[e];
  const int c  = (int)col[e];
  const float w = norm[e];
  atomicAdd(&out[(size_t)c * HID + ch], w * hw[(size_t)r * HID + ch]);
}

// self-loop contribution (norm = dinv^2) + bias + leaky-relu, fused
__global__ void k_selfloop_bias_act(const float* __restrict__ hw,
                                    const float* __restrict__ dinv,
                                    const float* __restrict__ bias,
                                    float* __restrict__ out) {
  const int idx = blockIdx.x * blockDim.x + threadIdx.x;   // NN*HID
  const int i   = idx >> 8;
  const int ch  = idx & (HID - 1);
  const float d = dinv[i];
  out[idx] = lrelu(out[idx] + d * d * hw[idx] + bias[ch]);
}

// flat[N,316] = concat(h2, x) — identical memory to the [B, 22*316] reshape
__global__ void k_build_flat(const float* __restrict__ h,
                             const float* __restrict__ x,
                             float* __restrict__ flat) {
  const int idx  = blockIdx.x * blockDim.x + threadIdx.x;  // NN*CAT
  const int node = idx / CAT;
  const int j    = idx - node * CAT;
  flat[idx] = (j < HID) ? h[(size_t)node * HID + j]
                        : x[(size_t)node * FEA + (j - HID)];
}

// final head: out[B,4] = z @ Wo + bo   (4 MFLOP -> scalar kernel, no tail mask)
__global__ void k_out_head(const float* __restrict__ z, const float* __restrict__ Wo,
                           const float* __restrict__ bo, float* __restrict__ out) {
  const int idx = blockIdx.x * blockDim.x + threadIdx.x;   // BATCH*4
  const int b = idx >> 2, o = idx & 3;
  float s = bo[o];
  const float* __restrict__ zr = z + (size_t)b * HID;
  for (int k = 0; k < HID; ++k) s += zr[k] * Wo[k * 4 + o];
  out[idx] = s;
}

extern "C" void kernel_launch(void* const* d_in, const int* in_sizes, int n_in,
                              void* d_out, int out_size, void* d_ws, size_t ws_size,
                              hipStream_t stream) {
  (void)in_sizes; (void)n_in; (void)out_size; (void)ws_size;
  const float*     x   = (const float*)d_in[0];
  const long long* ei  = (const long long*)d_in[1];   // int64 [2,E]
  const float*     ew  = (const float*)d_in[2];
  const float*     W1  = (const float*)d_in[3];
  const float*     b1  = (const float*)d_in[4];
  const float*     W2  = (const float*)d_in[5];
  const float*     b2  = (const float*)d_in[6];
  const float*     Wf0 = (const float*)d_in[7];
  const float*     bf0 = (const float*)d_in[8];
  const float*     Wf1 = (const float*)d_in[9];
  const float*     bf1 = (const float*)d_in[10];
  const float*     Wf2 = (const float*)d_in[11];
  const float*     bf2 = (const float*)d_in[12];
  const float*     Wo  = (const float*)d_in[13];
  const float*     bo  = (const float*)d_in[14];
  float*           out = (float*)d_out;

  const long long* row = ei;
  const long long* col = ei + EE;

  // workspace layout (floats): ~154 MB total; flat overlays dead hw/h1 region
  float* ws   = (float*)d_ws;
  float* deg  = ws;                                  // NN (becomes dinv)
  float* norm = deg + NN;                            // EE
  float* hw   = norm + EE;                           // NN*HID
  float* h1   = hw + (size_t)NN * HID;               // NN*HID
  float* h2   = h1 + (size_t)NN * HID;               // NN*HID
  float* z0   = h2 + (size_t)NN * HID;               // BATCH*HID
  float* z1   = z0 + (size_t)BATCH * HID;            // BATCH*HID
  float* W1p  = z1 + (size_t)BATCH * HID;            // FEA*HID
  float* W2p  = W1p + (size_t)FEA * HID;             // HID*HID
  float* Wf0p = W2p + (size_t)HID * HID;             // MLP_IN*HID
  float* Wf1p = Wf0p + (size_t)MLP_IN * HID;         // HID*HID
  float* Wf2p = Wf1p + (size_t)HID * HID;            // HID*HID
  float* flat = hw;  // NN*CAT floats fit inside hw+h1 (2*NN*HID), both dead

  // --- weight packing (depends only on inputs) ---
  k_pack_w<<<(FEA    * HID + 255) / 256, 256, 0, stream>>>(W1,  W1p,  FEA);
  k_pack_w<<<(HID    * HID + 255) / 256, 256, 0, stream>>>(W2,  W2p,  HID);
  k_pack_w<<<(MLP_IN * HID + 255) / 256, 256, 0, stream>>>(Wf0, Wf0p, MLP_IN);
  k_pack_w<<<(HID    * HID + 255) / 256, 256, 0, stream>>>(Wf1, Wf1p, HID);
  k_pack_w<<<(HID    * HID + 255) / 256, 256, 0, stream>>>(Wf2, Wf2p, HID);

  // --- gcn_norm ---
  k_init_deg <<<(NN + 255) / 256, 256, 0, stream>>>(deg);
  k_deg_accum<<<(EE + 255) / 256, 256, 0, stream>>>(col, ew, deg);
  k_dinv     <<<(NN + 255) / 256, 256, 0, stream>>>(deg);
  k_norm     <<<(EE + 255) / 256, 256, 0, stream>>>(row, col, ew, deg, norm);

  const dim3 gBig(NN / 128, HID / 64);           // (352, 4)
  const dim3 gMlp(BATCH / 128, HID / 64);        // (16, 4)
  const int  n4 = NN * HID / 4;

  // --- conv1: hw = x@W1 ; h1 = lrelu(scatter + selfloop + b1) ---
  k_gemm_wmma<false, false><<<gBig, 256, 0, stream>>>(x, W1p, nullptr, hw, FEA);
  k_zero<<<(n4 + 255) / 256, 256, 0, stream>>>((float4*)h1, n4);
  k_scatter<<<EE, HID, 0, stream>>>(hw, row, col, norm, h1);
  k_selfloop_bias_act<<<NN * HID / 256, 256, 0, stream>>>(hw, deg, b1, h1);

  // --- conv2: hw = h1@W2 ; h2 = lrelu(scatter + selfloop + b2) ---
  k_gemm_wmma<false, false><<<gBig, 256, 0, stream>>>(h1, W2p, nullptr, hw, HID);
  k_zero<<<(n4 + 255) / 256, 256, 0, stream>>>((float4*)h2, n4);
  k_scatter<<<EE, HID, 0, stream>>>(hw, row, col, norm, h2);
  k_selfloop_bias_act<<<NN * HID / 256, 256, 0, stream>>>(hw, deg, b2, h2);

  // --- MLP head ---
  k_build_flat<<<NN * CAT / 256, 256, 0, stream>>>(h2, x, flat);
  k_gemm_wmma<true, true ><<<gMlp, 256, 0, stream>>>(flat, Wf0p, bf0, z0, MLP_IN);
  k_gemm_wmma<true, true ><<<gMlp, 256, 0, stream>>>(z0,   Wf1p, bf1, z1, HID);
  k_gemm_wmma<true, true ><<<gMlp, 256, 0, stream>>>(z1,   Wf2p, bf2, z0, HID);
  k_out_head<<<BATCH * 4 / 256, 256, 0, stream>>>(z0, Wo, bo, out);
}